// Involution2d_84825604096329
// MI455X (gfx1250) — compile-verified
//
#include <hip/hip_runtime.h>
#include <hip/hip_bf16.h>

typedef __attribute__((ext_vector_type(2))) float v2f;
typedef __attribute__((ext_vector_type(8))) float v8f;
typedef __attribute__((ext_vector_type(4))) unsigned int u32x4;
typedef __attribute__((ext_vector_type(8))) int i32x8;
typedef __attribute__((ext_vector_type(4))) int i32x4;

// Problem constants (from reference)
#define BATCH 4
#define CIN   256      // C
#define HW    4096     // 64*64
#define WDIM  64
#define NOUT  784      // G*K*K = 16*49
#define GRP   16
#define KK    49

// GEMM tiling
#define MTILE 128      // pixels per workgroup
#define NTILE 112      // 7 x 16 outputs per workgroup
#define KCH   16       // K chunk staged in LDS
#define LDA_S 144      // A LDS pitch: 128 + 16 pad (TDM pad: interval 128 dw, amount 16 dw)
#define LDB_S 18       // B LDS pitch: 16 k-values + 2 pad (k-pair contiguous layout)

// ---------------------------------------------------------------------------
// Kernel A: kern[b, p, n] = sum_c x[b, c, p] * w[n, c] + bias[n]
// GEMM on the fp32 matrix pipe: V_WMMA_F32_16X16X4_F32.
// A tile (16 x 128 fp32, row stride HW) staged by the Tensor Data Mover with
// LDS padding to pitch 144; B tile staged as [n][k] pairs for b64 fragments.
// Grid: (7 N-blocks, 32 M-blocks, 4 batches), Block: 256 threads (8 waves).
// ---------------------------------------------------------------------------
__global__ __launch_bounds__(256)
void involution_kern_gemm(const float* __restrict__ x,
                          const float* __restrict__ w,
                          const float* __restrict__ bias,
                          float* __restrict__ kern_ws) {
    __shared__ __align__(16) float lds_A[KCH * LDA_S];    // [k][pixel], pitch 144: 9216 B
    __shared__ __align__(16) float lds_B[NTILE * LDB_S];  // [n][k],     pitch 18:  8064 B

    const int t     = threadIdx.x;
    const int wv    = t >> 5;          // wave id 0..7
    const int lane  = t & 31;
    const int ll    = lane & 15;       // low lane index
    const int lh    = lane >> 4;       // lane half (0/1)

    const int nbase = blockIdx.x * NTILE;      // 0..672
    const int p0    = blockIdx.y * MTILE;      // pixel block base
    const int b     = blockIdx.z;

    const float* xb = x + (size_t)b * CIN * HW + p0;   // tile origin (pixel p0, channel 0)

    // ---- TDM descriptor invariants (group1 + LDS addr) ----
    const unsigned int lds_a_off = (unsigned int)(uintptr_t)&lds_A[0]; // LDS byte offset
    // group1: data_size=4B(2), pad_enable, pad_interval=128 dw (code 6),
    //         pad_amount=16 dw (code 15); tensor_dim0=4096, tensor_dim1=256,
    //         tile_dim0=128, tile_dim1=16, tensor_dim0_stride=4096.
    i32x8 g1;
    g1[0] = (int)((2u << 16) | (1u << 20) | (6u << 22) | (15u << 25));
    g1[1] = (int)((4096u & 0xFFFFu) << 16);            // tensor_dim0[15:0] << 16
    g1[2] = (int)((4096u >> 16) | ((256u & 0xFFFFu) << 16)); // dim0[31:16] | dim1[15:0]<<16
    g1[3] = (int)((256u >> 16) | (128u << 16));        // dim1[31:16] | tile_dim0=128
    g1[4] = (int)(16u);                                // tile_dim1=16, tile_dim2=0
    g1[5] = (int)(4096u);                              // tensor_dim0_stride[31:0]
    g1[6] = 0;                                         // stride0[47:32] | stride1[15:0]
    g1[7] = 0;
    const i32x4 gz4 = (i32x4)0;                        // groups 2/3 unused (2D tile)
    const i32x8 gz8 = (i32x8)0;                        // trailing descriptor group unused

    v8f acc[7];
#pragma unroll
    for (int nt = 0; nt < 7; ++nt) acc[nt] = (v8f)0.0f;

#pragma unroll 1
    for (int kk = 0; kk < CIN; kk += KCH) {
        // ---- stage A via Tensor Data Mover: 16 rows x 128 pixels, pad->pitch 144 ----
        if (wv == 0) {
            const unsigned long long ga =
                (unsigned long long)(uintptr_t)(xb + (size_t)kk * HW);
            u32x4 g0;
            g0[0] = 1u;                                        // count=1, normal mode
            g0[1] = lds_a_off;                                 // lds_addr
            g0[2] = (unsigned int)ga;                          // global_addr[31:0]
            g0[3] = (unsigned int)((ga >> 32) & 0x1FFFFFFu) | (2u << 30); // [56:32] | type=2
            __builtin_amdgcn_tensor_load_to_lds(g0, g1, gz4, gz4, gz8, 0);
            __builtin_amdgcn_s_wait_tensorcnt(0);
        }
        // ---- stage B: w[nbase+n, kk+2jp..+1] -> lds_B[n][2jp] (float2, no transpose) ----
#pragma unroll
        for (int i = 0; i < 4; ++i) {
            int e = t + i * 256;                       // 112 n * 8 k-pairs = 896
            if (e < NTILE * (KCH / 2)) {
                int n  = e >> 3;
                int jp = e & 7;
                const float2 v = *(const float2*)&w[(size_t)(nbase + n) * CIN + kk + 2 * jp];
                *(float2*)&lds_B[n * LDB_S + 2 * jp] = v;
            }
        }
        // prefetch next B chunk (global_prefetch_b8 path)
        if (kk + KCH < CIN && (t >> 1) < NTILE) {
            __builtin_prefetch(&w[(size_t)(nbase + (t >> 1)) * CIN + kk + KCH], 0, 2);
        }
        __syncthreads();

        // ---- compute: 4 K-steps of 4, 7 N tiles, WMMA f32 16x16x4 ----
#pragma unroll
        for (int j4 = 0; j4 < 4; ++j4) {
            const int kl = j4 * 4 + 2 * lh;            // K for VGPR0 (ISA A/B lane layout)
            v2f a;
            a.x = lds_A[kl * LDA_S + wv * 16 + ll];
            a.y = lds_A[(kl + 1) * LDA_S + wv * 16 + ll];
#pragma unroll
            for (int nt = 0; nt < 7; ++nt) {
                const v2f bb = *(const v2f*)&lds_B[(nt * 16 + ll) * LDB_S + kl]; // b64
                acc[nt] = __builtin_amdgcn_wmma_f32_16x16x4_f32(
                    false, a, false, bb, (short)0, acc[nt], false, false);
            }
        }
        __syncthreads();
    }

    // ---- epilogue: add bias, store kern_ws[b][p][n] ----
    const int pb = p0 + wv * 16 + 8 * lh;              // base pixel for this lane-half
#pragma unroll
    for (int nt = 0; nt < 7; ++nt) {
        const int n = nbase + nt * 16 + ll;
        const float bv = bias[n];
#pragma unroll
        for (int r = 0; r < 8; ++r) {
            // D layout: VGPR r -> row r (lanes 0-15) / row r+8 (lanes 16-31)
            kern_ws[(size_t)(b * HW + pb + r) * NOUT + n] = acc[nt][r] + bv;
        }
    }
}

// ---------------------------------------------------------------------------
// Kernel B: out[b,c,h,w] = sum_{kh,kw} x[b,c,h+kh-3,w+kw-3] * kern[b,p,g*49+kh*7+kw]
// Grid: (16 pixel tiles, 16 groups, 4 batches), Block: 256 threads.
// ---------------------------------------------------------------------------
#define HS  22          // halo side (16 + 6)
#define HP  24          // padded halo pitch
#define CH_LDS (HS * HP)

__global__ __launch_bounds__(256)
void involution_gather(const float* __restrict__ x,
                       const float* __restrict__ kern_ws,
                       float* __restrict__ out) {
    __shared__ float lx[GRP * CH_LDS];   // 16 ch x 22x24 = 33792 B

    const int t    = threadIdx.x;
    const int tile = blockIdx.x;
    const int g    = blockIdx.y;
    const int b    = blockIdx.z;
    const int th0  = (tile >> 2) * 16;
    const int tw0  = (tile & 3) * 16;

    // ---- stage 16-channel 22x22 halo (zero padded) ----
    for (int e = t; e < GRP * HS * HS; e += 256) {
        int c   = e / (HS * HS);
        int rem = e - c * (HS * HS);
        int hy  = rem / HS;
        int hx  = rem - hy * HS;
        int gh  = th0 + hy - 3;
        int gw  = tw0 + hx - 3;
        float v = 0.0f;
        if ((unsigned)gh < (unsigned)WDIM && (unsigned)gw < (unsigned)WDIM) {
            v = x[((size_t)(b * CIN + g * GRP + c)) * HW + gh * WDIM + gw];
        }
        lx[c * CH_LDS + hy * HP + hx] = v;
    }
    __syncthreads();

    // ---- each thread owns one pixel; 49 kern taps in registers ----
    const int py = t >> 4;
    const int px = t & 15;
    const int p  = (th0 + py) * WDIM + (tw0 + px);

    const float* kp = kern_ws + ((size_t)(b * HW + p)) * NOUT + g * KK;
    float kq[KK];
#pragma unroll
    for (int k = 0; k < KK; ++k) kq[k] = kp[k];

#pragma unroll
    for (int c = 0; c < GRP; ++c) {
        const float* lc = &lx[c * CH_LDS];
        float a = 0.0f;
#pragma unroll
        for (int kh = 0; kh < 7; ++kh) {
#pragma unroll
            for (int kw = 0; kw < 7; ++kw) {
                a = fmaf(lc[(py + kh) * HP + (px + kw)], kq[kh * 7 + kw], a);
            }
        }
        out[((size_t)(b * CIN + g * GRP + c)) * HW + p] = a;
    }
}

// ---------------------------------------------------------------------------
extern "C" void kernel_launch(void* const* d_in, const int* in_sizes, int n_in,
                              void* d_out, int out_size, void* d_ws, size_t ws_size,
                              hipStream_t stream) {
    (void)in_sizes; (void)n_in; (void)out_size; (void)ws_size;
    const float* x    = (const float*)d_in[0];   // [4,256,64,64]
    const float* wk   = (const float*)d_in[1];   // [784,256]
    const float* bk   = (const float*)d_in[2];   // [784]
    float* out        = (float*)d_out;           // [4,256,64,64]
    float* kern_ws    = (float*)d_ws;            // [4,4096,784] = 51.4 MB

    dim3 gridA(NOUT / NTILE, HW / MTILE, BATCH); // (7, 32, 4)
    involution_kern_gemm<<<gridA, 256, 0, stream>>>(x, wk, bk, kern_ws);

    dim3 gridB(16, GRP, BATCH);                  // (16 tiles, 16 groups, 4)
    involution_gather<<<gridB, 256, 0, stream>>>(x, kern_ws, out);
}